// AnchorSegmentMixer_55559696941464
// MI455X (gfx1250) — compile-verified
//
#include <hip/hip_runtime.h>
#include <stdint.h>
#include <math.h>

// ---------------------------------------------------------------------------
// AnchorSegmentMixer for MI455X (gfx1250).
// B=256 rows, S=160000 samples, MIX_NUM=2  =>  one roll-by-1 mix.
//   energy[n]   = mean(w[n]^2)
//   ratio[n]    = clip(sqrt(energy[n]/max(energy[n+1], 1e-10)), 0.02, 50)
//   mixtures[n] = w[n] + ratio[n]*w[n+1 mod B]
//   targets     = w
// Memory-bound (~0.25 flop/byte): b128 loads with 4-deep per-wave MLP,
// async global->LDS staging (ASYNCcnt path), non-temporal b128 stores.
// ---------------------------------------------------------------------------

typedef float v4f __attribute__((ext_vector_type(4)));
typedef int   v4i __attribute__((ext_vector_type(4)));

#define NROWS   256
#define NSAMP   160000
#define NV4     (NSAMP / 4)        // 40000 float4 per row
#define ECHUNKS 4
#define EV4     (NV4 / ECHUNKS)    // 10000 float4 per energy chunk
#define MIX_EPS 1e-10f

#define CHUNK_V4 1024              // float4 per mix block (16 KB LDS stage)
#define NCHUNKS  ((NV4 + CHUNK_V4 - 1) / CHUNK_V4)   // 40

#if __has_builtin(__builtin_amdgcn_global_load_async_to_lds_b128)
#define HAVE_ASYNC_LDS 1
#endif

#if defined(HAVE_ASYNC_LDS)
// Builtin signature (per hipcc diagnostic): (int4 AS1*, int4 AS3*, imm, imm)
typedef __attribute__((address_space(1))) v4i* gv4i_p;
typedef __attribute__((address_space(3))) v4i* lv4i_p;

__device__ __forceinline__ void wait_asynccnt0() {
#if __has_builtin(__builtin_amdgcn_s_wait_asynccnt)
  __builtin_amdgcn_s_wait_asynccnt(0);
#else
  asm volatile("s_wait_asynccnt 0" ::: "memory");
#endif
}
#endif

// ---- Mix + copy (placed first so the disasm snippet shows the async path) --
__global__ __launch_bounds__(256) void mix_kernel(
    const float* __restrict__ w, const float* __restrict__ energy,
    float* __restrict__ out) {
  const int row  = blockIdx.y;
  const int nrow = (row + 1) & (NROWS - 1);

  const float e0 = energy[row];
  const float e1 = energy[nrow];
  float ratio = sqrtf(e0 / fmaxf(e1, MIX_EPS));
  ratio = fminf(fmaxf(ratio, 0.02f), 50.0f);

  const v4f* __restrict__ cur = (const v4f*)(w + (size_t)row  * NSAMP);
  const v4f* __restrict__ nxt = (const v4f*)(w + (size_t)nrow * NSAMP);
  v4f* __restrict__ mix = (v4f*)(out + (size_t)row * NSAMP);
  v4f* __restrict__ tgt = (v4f*)(out + (size_t)NROWS * NSAMP + (size_t)row * NSAMP);

  const int base = blockIdx.x * CHUNK_V4;
  const int tid  = threadIdx.x;

#if defined(HAVE_ASYNC_LDS)
  __shared__ v4f stage[CHUNK_V4];  // 16 KB of the 320 KB/WGP LDS

  // Stage next-row tile into LDS via the CDNA5 async path (ASYNCcnt).
#pragma unroll
  for (int k = 0; k < 4; ++k) {
    const int j = base + tid + k * 256;
    if (j < NV4) {
      __builtin_amdgcn_global_load_async_to_lds_b128(
          (gv4i_p)(&nxt[j]), (lv4i_p)(&stage[tid + k * 256]),
          /*offset=*/0, /*cpol=*/0);
    }
  }
#endif

  // Current-row tile straight to VGPRs; overlaps with the async transfers.
  v4f c[4];
#pragma unroll
  for (int k = 0; k < 4; ++k) {
    const int j = base + tid + k * 256;
    if (j < NV4) c[k] = cur[j];
  }

#if defined(HAVE_ASYNC_LDS)
  wait_asynccnt0();
  __syncthreads();
#endif

#pragma unroll
  for (int k = 0; k < 4; ++k) {
    const int j = base + tid + k * 256;
    if (j < NV4) {
#if defined(HAVE_ASYNC_LDS)
      v4f n = stage[tid + k * 256];   // ds_load_b128
#else
      v4f n = nxt[j];
#endif
      v4f m;
      m.x = fmaf(ratio, n.x, c[k].x);
      m.y = fmaf(ratio, n.y, c[k].y);
      m.z = fmaf(ratio, n.z, c[k].z);
      m.w = fmaf(ratio, n.w, c[k].w);
      // Outputs are never re-read: non-temporal b128 stores keep the 328 MB
      // of writes from evicting the 164 MB input out of the 192 MB L2.
      __builtin_nontemporal_store(m,    &mix[j]);
      __builtin_nontemporal_store(c[k], &tgt[j]);
    }
  }
}

// ---- Pass 1: partial sum-of-squares per (row, chunk) -----------------------
// Manually unrolled x4 with independent accumulators: keeps 4 b128 loads in
// flight per wave instead of the 1-load / s_wait_loadcnt-0 serial loop the
// compiler produced from the naive form.
__global__ __launch_bounds__(256) void energy_partial_kernel(
    const float* __restrict__ w, float* __restrict__ partial) {
  const int row   = blockIdx.y;
  const int chunk = blockIdx.x;
  const v4f* __restrict__ wr = (const v4f*)(w + (size_t)row * NSAMP);

  const int jend = (chunk + 1) * EV4;
  int j = chunk * EV4 + threadIdx.x;

  float a0 = 0.0f, a1 = 0.0f, a2 = 0.0f, a3 = 0.0f;
  for (; j + 768 < jend; j += 1024) {
    v4f v0 = wr[j];
    v4f v1 = wr[j + 256];
    v4f v2 = wr[j + 512];
    v4f v3 = wr[j + 768];
    a0 = fmaf(v0.x, v0.x, a0); a0 = fmaf(v0.y, v0.y, a0);
    a0 = fmaf(v0.z, v0.z, a0); a0 = fmaf(v0.w, v0.w, a0);
    a1 = fmaf(v1.x, v1.x, a1); a1 = fmaf(v1.y, v1.y, a1);
    a1 = fmaf(v1.z, v1.z, a1); a1 = fmaf(v1.w, v1.w, a1);
    a2 = fmaf(v2.x, v2.x, a2); a2 = fmaf(v2.y, v2.y, a2);
    a2 = fmaf(v2.z, v2.z, a2); a2 = fmaf(v2.w, v2.w, a2);
    a3 = fmaf(v3.x, v3.x, a3); a3 = fmaf(v3.y, v3.y, a3);
    a3 = fmaf(v3.z, v3.z, a3); a3 = fmaf(v3.w, v3.w, a3);
  }
  for (; j < jend; j += 256) {   // remainder (EV4 = 39*256 + 16)
    v4f v = wr[j];
    a0 = fmaf(v.x, v.x, a0); a0 = fmaf(v.y, v.y, a0);
    a0 = fmaf(v.z, v.z, a0); a0 = fmaf(v.w, v.w, a0);
  }
  float acc = (a0 + a1) + (a2 + a3);

  __shared__ float red[256];
  red[threadIdx.x] = acc;
  __syncthreads();
#pragma unroll
  for (int s = 128; s > 0; s >>= 1) {
    if (threadIdx.x < s) red[threadIdx.x] += red[threadIdx.x + s];
    __syncthreads();
  }
  if (threadIdx.x == 0) partial[row * ECHUNKS + chunk] = red[0];
}

// ---- Pass 2: finalize energies --------------------------------------------
__global__ __launch_bounds__(256) void energy_final_kernel(
    const float* __restrict__ partial, float* __restrict__ energy) {
  const int t = threadIdx.x;  // one thread per row
  float s = partial[t * ECHUNKS + 0] + partial[t * ECHUNKS + 1] +
            partial[t * ECHUNKS + 2] + partial[t * ECHUNKS + 3];
  energy[t] = s * (1.0f / (float)NSAMP);
}

// ---------------------------------------------------------------------------
extern "C" void kernel_launch(void* const* d_in, const int* in_sizes, int n_in,
                              void* d_out, int out_size, void* d_ws, size_t ws_size,
                              hipStream_t stream) {
  (void)in_sizes; (void)n_in; (void)out_size; (void)ws_size;
  const float* w  = (const float*)d_in[0];
  float* out      = (float*)d_out;
  float* partial  = (float*)d_ws;              // 1024 floats
  float* energy   = partial + NROWS * ECHUNKS; // 256 floats

  energy_partial_kernel<<<dim3(ECHUNKS, NROWS), 256, 0, stream>>>(w, partial);
  energy_final_kernel<<<1, 256, 0, stream>>>(partial, energy);
  mix_kernel<<<dim3(NCHUNKS, NROWS), 256, 0, stream>>>(w, energy, out);
}